// NestedGraphTitanNL_4303557230752
// MI455X (gfx1250) — compile-verified
//
#include <hip/hip_runtime.h>
#include <math.h>

typedef __attribute__((ext_vector_type(2))) float v2f;
typedef __attribute__((ext_vector_type(8))) float v8f;

#define DD 128
#define ROWS 8192   // 32*64*4 rows of the row-major activation matrix
#define SEQ 64
#define NSEQ 128    // ROWS / SEQ sequences (raw reshape => consecutive row blocks)

__device__ __forceinline__ float sigmoidf_(float x){ return 1.f/(1.f+expf(-x)); }
__device__ __forceinline__ float siluf_(float x){ return x*sigmoidf_(x); }

// blockDim.x must be 128; all 128 threads must call.
__device__ __forceinline__ float blockReduceSum128(float v, float* s){
  int t = threadIdx.x;
  s[t] = v; __syncthreads();
  #pragma unroll
  for (int off = 64; off > 0; off >>= 1){
    if (t < off) s[t] += s[t + off];
    __syncthreads();
  }
  float r = s[0];
  __syncthreads();
  return r;
}

// ---------------------------------------------------------------------------
// WMMA FP32 GEMM: C[M,N] = act(A[M,K] @ B[K,N] + bias)
// One wave (32 threads) per 16x16 output tile, K stepped by 4 via
// v_wmma_f32_16x16x4_f32. M,N multiples of 16; K multiple of 4.
// ---------------------------------------------------------------------------
template<int ACT>   // 0 = none, 1 = silu
__global__ void gemm_wmma_f32(const float* __restrict__ A, const float* __restrict__ B,
                              const float* __restrict__ bias, float* __restrict__ C,
                              int M, int N, int K){
  (void)M;
  const int lane = threadIdx.x;
  const int hi   = lane >> 4;     // 0 or 1
  const int lo   = lane & 15;
  const int arow = blockIdx.y * 16 + lo;   // A-frag row (M)
  const int col  = blockIdx.x * 16 + lo;   // B-frag / C col (N)
  const float* Ar = A + (size_t)arow * K;

  v8f acc = {0.f,0.f,0.f,0.f,0.f,0.f,0.f,0.f};
  for (int k0 = 0; k0 < K; k0 += 4){
    const int ka = k0 + 2*hi;
    v2f a, b;
    a.x = Ar[ka];
    a.y = Ar[ka + 1];
    b.x = B[(size_t)ka      * N + col];
    b.y = B[(size_t)(ka + 1)* N + col];
    acc = __builtin_amdgcn_wmma_f32_16x16x4_f32(false, a, false, b,
                                                (short)0, acc, false, false);
  }
  const float bv = bias ? bias[col] : 0.f;
  #pragma unroll
  for (int r = 0; r < 8; r++){
    const int m = blockIdx.y * 16 + r + 8*hi;
    float v = acc[r] + bv;
    if (ACT == 1) v = siluf_(v);
    C[(size_t)m * N + col] = v;
  }
}

// ---------------------------------------------------------------------------
// Embedding: h = LN(x @ emb_W + emb_b) + pos_emb[s];  grid=ROWS, block=128
// ---------------------------------------------------------------------------
__global__ void embed_kernel(const float* __restrict__ x, const float* __restrict__ W,
                             const float* __restrict__ bb, const float* __restrict__ g,
                             const float* __restrict__ be, const float* __restrict__ pe,
                             float* __restrict__ h){
  __shared__ float sred[128];
  const int r = blockIdx.x;
  const int d = threadIdx.x;
  const float* xr = x + (size_t)r * 34;
  float acc = bb[d];
  #pragma unroll
  for (int f = 0; f < 34; f++) acc += xr[f] * W[f*DD + d];
  float m   = blockReduceSum128(acc, sred) * (1.f/128.f);
  float c   = acc - m;
  float var = blockReduceSum128(c*c, sred) * (1.f/128.f);
  float y   = g[d] * c * rsqrtf(var + 1e-5f) + be[d];
  const int s = (r >> 2) & 63;                 // row = ((b*64+s)*4+n)
  h[(size_t)r*DD + d] = y + pe[s*DD + d];
}

// ---------------------------------------------------------------------------
// out = LN(x + resid) (resid may be null);  grid=ROWS, block=128
// ---------------------------------------------------------------------------
__global__ void add_ln_kernel(const float* __restrict__ x, const float* __restrict__ resid,
                              const float* __restrict__ g, const float* __restrict__ be,
                              float* __restrict__ out){
  __shared__ float sred[128];
  const int r = blockIdx.x;
  const int d = threadIdx.x;
  float v = x[(size_t)r*DD + d] + (resid ? resid[(size_t)r*DD + d] : 0.f);
  float m   = blockReduceSum128(v, sred) * (1.f/128.f);
  float c   = v - m;
  float var = blockReduceSum128(c*c, sred) * (1.f/128.f);
  out[(size_t)r*DD + d] = g[d] * c * rsqrtf(var + 1e-5f) + be[d];
}

// ---------------------------------------------------------------------------
// CMS level accumulate: agg (+)= softmax(lw)[level] * LN(t2 + h)
// ---------------------------------------------------------------------------
__global__ void cms_accum_kernel(const float* __restrict__ t2, const float* __restrict__ h,
                                 const float* __restrict__ g, const float* __restrict__ be,
                                 const float* __restrict__ lw, int level,
                                 float* __restrict__ agg){
  __shared__ float sred[128];
  const int r = blockIdx.x;
  const int d = threadIdx.x;
  float v = t2[(size_t)r*DD + d] + h[(size_t)r*DD + d];
  float m   = blockReduceSum128(v, sred) * (1.f/128.f);
  float c   = v - m;
  float var = blockReduceSum128(c*c, sred) * (1.f/128.f);
  float y   = g[d] * c * rsqrtf(var + 1e-5f) + be[d];
  float w0 = lw[0], w1 = lw[1], w2 = lw[2];
  float mx = fmaxf(w0, fmaxf(w1, w2));
  float e0 = expf(w0-mx), e1 = expf(w1-mx), e2 = expf(w2-mx);
  float wsum = e0 + e1 + e2;
  float w = ((level==0) ? e0 : (level==1) ? e1 : e2) / wsum;
  float val = w * y;
  size_t o = (size_t)r*DD + d;
  if (level == 0) agg[o] = val; else agg[o] += val;
}

// ---------------------------------------------------------------------------
// eta/alpha output heads: 32->1 dot + scaled sigmoid.  One thread per row.
// ---------------------------------------------------------------------------
__global__ void head_kernel(const float* __restrict__ ehid, const float* __restrict__ ahid,
                            const float* __restrict__ eW2, const float* __restrict__ eb2,
                            const float* __restrict__ aW2, const float* __restrict__ ab2,
                            float* __restrict__ eta, float* __restrict__ alpha){
  const int r = blockIdx.x * blockDim.x + threadIdx.x;
  if (r >= ROWS) return;
  float se = eb2[0], sa = ab2[0];
  const float* er = ehid + (size_t)r*32;
  const float* ar = ahid + (size_t)r*32;
  #pragma unroll
  for (int j = 0; j < 32; j++){ se += er[j]*eW2[j]; sa += ar[j]*aW2[j]; }
  eta[r]   = sigmoidf_(se) * 0.1f + 0.01f;
  alpha[r] = sigmoidf_(sa) * 0.5f + 0.5f;
}

// ---------------------------------------------------------------------------
// Delta-memory scan. One workgroup (256 threads) per sequence; state M
// (128x128 fp32) lives in VGPRs: thread t owns row (t>>1), cols (t&1)*64..+63.
//   out_t = M q_t ; kn = k/max(||k||,1e-12); Mk = M kn;
//   M = alpha*M + eta*(v - Mk) kn^T
// ---------------------------------------------------------------------------
__global__ void delta_scan_kernel(const float* __restrict__ q, const float* __restrict__ k,
                                  const float* __restrict__ vhat,
                                  const float* __restrict__ eta, const float* __restrict__ alpha,
                                  float* __restrict__ out){
  __shared__ float qs[128], kn[128], vs[128], mk[128], ov[128];
  __shared__ float redq[256], redk[256];
  __shared__ float sInv;
  const int p = blockIdx.x;
  const int t = threadIdx.x;
  const int row   = t >> 1;
  const int jbase = (t & 1) * 64;
  float Mreg[64];
  #pragma unroll
  for (int j = 0; j < 64; j++) Mreg[j] = 0.f;

  for (int step = 0; step < SEQ; step++){
    const int r = p * SEQ + step;
    if (t < 128){
      qs[t] = q[(size_t)r*DD + t];
      float kv = k[(size_t)r*DD + t];
      kn[t] = kv;
      vs[t] = vhat[(size_t)r*DD + t];
      redq[t] = kv * kv;
    }
    __syncthreads();
    #pragma unroll
    for (int off = 64; off > 0; off >>= 1){
      if (t < off) redq[t] += redq[t + off];
      __syncthreads();
    }
    if (t == 0) sInv = 1.f / fmaxf(sqrtf(redq[0]), 1e-12f);
    __syncthreads();
    if (t < 128) kn[t] *= sInv;
    __syncthreads();

    // matvec partials from VGPR-resident M
    float pq = 0.f, pk = 0.f;
    #pragma unroll
    for (int j = 0; j < 64; j++){
      float mm = Mreg[j];
      pq = fmaf(mm, qs[jbase + j], pq);
      pk = fmaf(mm, kn[jbase + j], pk);
    }
    redq[t] = pq; redk[t] = pk;
    __syncthreads();
    if ((t & 1) == 0){
      int rr = t >> 1;
      ov[rr] = redq[t] + redq[t + 1];
      mk[rr] = redk[t] + redk[t + 1];
    }
    __syncthreads();
    if (t < 128) out[(size_t)r*DD + t] = ov[t];

    const float e = eta[r];
    const float a = alpha[r];
    const float coef = e * (vs[row] - mk[row]);
    #pragma unroll
    for (int j = 0; j < 64; j++)
      Mreg[j] = fmaf(a, Mreg[j], coef * kn[jbase + j]);
    __syncthreads();
  }
}

// ---------------------------------------------------------------------------
// Regime: state = mean of last 3 timesteps. grid=128 (b*4+n), block=128
// ---------------------------------------------------------------------------
__global__ void state_kernel(const float* __restrict__ h, float* __restrict__ state){
  const int br = blockIdx.x;
  const int d  = threadIdx.x;
  const int b = br >> 2, n = br & 3;
  float s = 0.f;
  #pragma unroll
  for (int si = 61; si < 64; si++){
    const int r = (b*64 + si)*4 + n;
    s += h[(size_t)r*DD + d];
  }
  state[(size_t)br*DD + d] = s * (1.f/3.f);
}

// gm = mean over n, gs = std (ddof=1) over n.  grid=32, block=128
__global__ void gmgs_kernel(const float* __restrict__ state,
                            float* __restrict__ gm, float* __restrict__ gs){
  const int b = blockIdx.x;
  const int d = threadIdx.x;
  float x0 = state[(size_t)(b*4+0)*DD + d];
  float x1 = state[(size_t)(b*4+1)*DD + d];
  float x2 = state[(size_t)(b*4+2)*DD + d];
  float x3 = state[(size_t)(b*4+3)*DD + d];
  float m = (x0+x1+x2+x3) * 0.25f;
  float v = ((x0-m)*(x0-m) + (x1-m)*(x1-m) + (x2-m)*(x2-m) + (x3-m)*(x3-m)) * (1.f/3.f);
  gm[(size_t)b*DD + d] = m;
  gs[(size_t)b*DD + d] = sqrtf(v);
}

// det MLP: probs = softmax(silu([state,gm] @ W1 + b1) @ W2 + b2). grid=128, block=128
__global__ void det_kernel(const float* __restrict__ state, const float* __restrict__ gm,
                           const float* __restrict__ W1, const float* __restrict__ b1,
                           const float* __restrict__ W2, const float* __restrict__ b2,
                           float* __restrict__ probs){
  __shared__ float hid[128];
  __shared__ float lg[3];
  const int row = blockIdx.x;
  const int b = row >> 2;
  const int t = threadIdx.x;
  const float* st = state + (size_t)row*DD;
  const float* g  = gm + (size_t)b*DD;
  float acc = b1[t];
  for (int i = 0; i < 128; i++) acc += st[i] * W1[i*128 + t];
  for (int i = 0; i < 128; i++) acc += g[i]  * W1[(128 + i)*128 + t];
  hid[t] = siluf_(acc);
  __syncthreads();
  if (t < 3){
    float o = b2[t];
    for (int j = 0; j < 128; j++) o += hid[j] * W2[j*3 + t];
    lg[t] = o;
  }
  __syncthreads();
  if (t < 3){
    float mx = fmaxf(lg[0], fmaxf(lg[1], lg[2]));
    float ssum = expf(lg[0]-mx) + expf(lg[1]-mx) + expf(lg[2]-mx);
    probs[row*3 + t] = expf(lg[t]-mx) / ssum;
  }
}

// gate: alpha = sigmoid(relu([state,gm,gs,probs] @ W1 + b1) @ W2 + b2). grid=128, block=64
__global__ void gate_kernel(const float* __restrict__ state, const float* __restrict__ gm,
                            const float* __restrict__ gs, const float* __restrict__ probs,
                            const float* __restrict__ W1, const float* __restrict__ b1,
                            const float* __restrict__ W2, const float* __restrict__ b2,
                            float* __restrict__ alphag){
  __shared__ float hid[64];
  const int row = blockIdx.x;
  const int b = row >> 2;
  const int t = threadIdx.x;
  const float* st = state + (size_t)row*DD;
  const float* g  = gm + (size_t)b*DD;
  const float* s  = gs + (size_t)b*DD;
  const float* pr = probs + row*3;
  float acc = b1[t];
  for (int i = 0; i < 128; i++) acc += st[i] * W1[i*64 + t];
  for (int i = 0; i < 128; i++) acc += g[i]  * W1[(128 + i)*64 + t];
  for (int i = 0; i < 128; i++) acc += s[i]  * W1[(256 + i)*64 + t];
  for (int i = 0; i < 3;   i++) acc += pr[i] * W1[(384 + i)*64 + t];
  hid[t] = fmaxf(acc, 0.f);
  __syncthreads();
  if (t == 0){
    float o = b2[0];
    for (int j = 0; j < 64; j++) o += hid[j] * W2[j];
    alphag[row] = sigmoidf_(o);
  }
}

// regime 4x4 attention + mixing + LN.  grid=32, block=128
__global__ void regime_attn_kernel(const float* __restrict__ Q, const float* __restrict__ K,
                                   const float* __restrict__ V, const float* __restrict__ state,
                                   const float* __restrict__ alphag,
                                   const float* __restrict__ g, const float* __restrict__ be,
                                   float* __restrict__ outp){
  __shared__ float Sc[16];
  __shared__ float At[16];
  __shared__ float sred[128];
  const int b = blockIdx.x;
  const int t = threadIdx.x;
  if (t < 16){
    const int i = t >> 2, j = t & 3;
    const float* qi = Q + (size_t)(b*4 + i)*DD;
    const float* kj = K + (size_t)(b*4 + j)*DD;
    float s = 0.f;
    for (int d = 0; d < 128; d++) s += qi[d] * kj[d];
    Sc[t] = s * rsqrtf(128.f);
  }
  __syncthreads();
  if (t < 4){
    const int i = t;
    float mx = fmaxf(fmaxf(Sc[i*4+0], Sc[i*4+1]), fmaxf(Sc[i*4+2], Sc[i*4+3]));
    float e0 = expf(Sc[i*4+0]-mx), e1 = expf(Sc[i*4+1]-mx);
    float e2 = expf(Sc[i*4+2]-mx), e3 = expf(Sc[i*4+3]-mx);
    float sum = e0 + e1 + e2 + e3;
    At[i*4+0] = e0/sum; At[i*4+1] = e1/sum; At[i*4+2] = e2/sum; At[i*4+3] = e3/sum;
  }
  __syncthreads();
  float o[4];
  #pragma unroll
  for (int i = 0; i < 4; i++){
    const float al = alphag[b*4 + i];
    float acc = 0.f;
    #pragma unroll
    for (int j = 0; j < 4; j++){
      float mij = al * ((i == j) ? 1.f : 0.f) + (1.f - al) * At[i*4 + j];
      acc += mij * V[(size_t)(b*4 + j)*DD + t];
    }
    o[i] = acc + state[(size_t)(b*4 + i)*DD + t];
  }
  #pragma unroll
  for (int i = 0; i < 4; i++){
    float m   = blockReduceSum128(o[i], sred) * (1.f/128.f);
    float c   = o[i] - m;
    float var = blockReduceSum128(c*c, sred) * (1.f/128.f);
    outp[(size_t)(b*4 + i)*DD + t] = g[t] * c * rsqrtf(var + 1e-5f) + be[t];
  }
}

// ---------------------------------------------------------------------------
extern "C" void kernel_launch(void* const* d_in, const int* in_sizes, int n_in,
                              void* d_out, int out_size, void* d_ws, size_t ws_size,
                              hipStream_t stream){
  (void)in_sizes; (void)n_in; (void)out_size; (void)ws_size;
  int idx = 0;
  auto P = [&]() -> const float* { return (const float*)d_in[idx++]; };

  // ---- unpack (JAX pytree order: dict keys sorted, capitals < lowercase) ----
  const float* x = P();                                   // x [32,64,4,34]
  // cms: final_ln{b,g}, level_weights, lns[3]{b,g}, mlps[3]{W1,W2,b1,b2}
  const float* cms_fln_b = P();
  const float* cms_fln_g = P();
  const float* cms_lw    = P();
  const float *cms_ln_b[3], *cms_ln_g[3];
  for (int l = 0; l < 3; l++){ cms_ln_b[l] = P(); cms_ln_g[l] = P(); }
  const float *cms_W1[3], *cms_W2[3], *cms_b1[3], *cms_b2[3];
  for (int l = 0; l < 3; l++){ cms_W1[l] = P(); cms_W2[l] = P(); cms_b1[l] = P(); cms_b2[l] = P(); }
  const float* emb_W   = P();
  const float* emb_b   = P();
  const float* in_ln_b = P();
  const float* in_ln_g = P();
  // layers[4]: Wk,Wo,Wq,Wv, alpha{W1,W2,b1,b2}, bo, eta{W1,W2,b1,b2}, ln{b,g}, vgen{W1,W2,b1,b2}
  struct Lay { const float *Wk,*Wo,*Wq,*Wv,*aW1,*aW2,*ab1,*ab2,*bo,*eW1,*eW2,*eb1,*eb2,*ln_b,*ln_g,*vW1,*vW2,*vb1,*vb2; } L[4];
  for (int l = 0; l < 4; l++){
    L[l].Wk = P(); L[l].Wo = P(); L[l].Wq = P(); L[l].Wv = P();
    L[l].aW1 = P(); L[l].aW2 = P(); L[l].ab1 = P(); L[l].ab2 = P();
    L[l].bo = P();
    L[l].eW1 = P(); L[l].eW2 = P(); L[l].eb1 = P(); L[l].eb2 = P();
    L[l].ln_b = P(); L[l].ln_g = P();
    L[l].vW1 = P(); L[l].vW2 = P(); L[l].vb1 = P(); L[l].vb2 = P();
  }
  const float* pos_emb = P();
  // regime: Wk,Wq,Wv,bk,bq,bv, det{W1,W2,b1,b2}, gate{W1,W2,b1,b2}, ln{b,g}
  const float* rWk = P(); const float* rWq = P(); const float* rWv = P();
  const float* rbk = P(); const float* rbq = P(); const float* rbv = P();
  const float* dW1 = P(); const float* dW2 = P(); const float* db1 = P(); const float* db2 = P();
  const float* gW1 = P(); const float* gW2 = P(); const float* gb1 = P(); const float* gb2 = P();
  const float* rln_b = P(); const float* rln_g = P();

  // ---- workspace layout ----
  float* ws = (float*)d_ws;
  const size_t RC = (size_t)ROWS * DD;        // 1M floats / 4MB
  float* h      = ws;                  // current activations
  float* hn     = ws + 1*RC;           // next / LN output
  float* qb     = ws + 2*RC;
  float* kb     = ws + 3*RC;
  float* vb     = ws + 4*RC;           // v, then scan output
  float* t5     = ws + 5*RC;           // vhid / wo / cms t2
  float* t6     = ws + 6*RC;           // vhat / cms agg
  float* hid512 = ws + 7*RC;           // [8192,512] (4*RC)
  float* ehid   = ws + 11*RC;          // [8192,32]
  float* ahid   = ehid + (size_t)ROWS*32;
  float* etab   = ahid + (size_t)ROWS*32;
  float* alphab = etab + ROWS;
  float* stateb = alphab + ROWS;       // [128,128]
  float* gmb    = stateb + 128*128;    // [32,128]
  float* gsb    = gmb + 32*128;
  float* probsb = gsb + 32*128;        // [128,3]
  float* alphag = probsb + 128*3;      // [128]
  float* Qb     = alphag + 128;        // [128,128]
  float* Kb     = Qb + 128*128;
  float* Vb     = Kb + 128*128;

  auto G = [&](const float* A, const float* B, const float* bias, float* C,
               int M, int N, int Kk, int act){
    dim3 grid(N/16, M/16), block(32);
    if (act) gemm_wmma_f32<1><<<grid, block, 0, stream>>>(A, B, bias, C, M, N, Kk);
    else     gemm_wmma_f32<0><<<grid, block, 0, stream>>>(A, B, bias, C, M, N, Kk);
  };

  // ---- embedding + input LN + positional embedding ----
  embed_kernel<<<ROWS, 128, 0, stream>>>(x, emb_W, emb_b, in_ln_g, in_ln_b, pos_emb, h);

  // ---- 4 delta-memory layers ----
  for (int l = 0; l < 4; l++){
    G(h,  L[l].Wq, nullptr,   qb, ROWS, DD, DD, 0);          // q = h @ Wq
    G(h,  L[l].Wk, nullptr,   kb, ROWS, DD, DD, 0);          // k = h @ Wk
    G(h,  L[l].Wv, nullptr,   vb, ROWS, DD, DD, 0);          // v = h @ Wv
    G(vb, L[l].vW1, L[l].vb1, t5, ROWS, DD, DD, 1);          // silu(v@W1+b1)
    G(t5, L[l].vW2, L[l].vb2, t6, ROWS, DD, DD, 0);          // vhat
    G(h,  L[l].eW1, L[l].eb1, ehid, ROWS, 32, DD, 1);        // eta hidden
    G(h,  L[l].aW1, L[l].ab1, ahid, ROWS, 32, DD, 1);        // alpha hidden
    head_kernel<<<ROWS/256, 256, 0, stream>>>(ehid, ahid, L[l].eW2, L[l].eb2,
                                              L[l].aW2, L[l].ab2, etab, alphab);
    delta_scan_kernel<<<NSEQ, 256, 0, stream>>>(qb, kb, t6, etab, alphab, vb); // sout -> vb
    G(vb, L[l].Wo, L[l].bo, t5, ROWS, DD, DD, 0);            // sout @ Wo + bo
    add_ln_kernel<<<ROWS, 128, 0, stream>>>(t5, h, L[l].ln_g, L[l].ln_b, hn);
    float* tmp = h; h = hn; hn = tmp;
  }

  // ---- CMS ----
  for (int l = 0; l < 3; l++){
    G(h,      cms_W1[l], cms_b1[l], hid512, ROWS, 512, DD, 1);
    G(hid512, cms_W2[l], cms_b2[l], t5,     ROWS, DD, 512, 0);
    cms_accum_kernel<<<ROWS, 128, 0, stream>>>(t5, h, cms_ln_g[l], cms_ln_b[l],
                                               cms_lw, l, t6);
  }
  add_ln_kernel<<<ROWS, 128, 0, stream>>>(t6, nullptr, cms_fln_g, cms_fln_b, hn);
  float* hc = hn;

  // ---- regime memory ----
  state_kernel<<<128, 128, 0, stream>>>(hc, stateb);
  gmgs_kernel<<<32, 128, 0, stream>>>(stateb, gmb, gsb);
  det_kernel<<<128, 128, 0, stream>>>(stateb, gmb, dW1, db1, dW2, db2, probsb);
  gate_kernel<<<128, 64, 0, stream>>>(stateb, gmb, gsb, probsb, gW1, gb1, gW2, gb2, alphag);
  G(stateb, rWq, rbq, Qb, 128, DD, DD, 0);
  G(stateb, rWk, rbk, Kb, 128, DD, DD, 0);
  G(stateb, rWv, rbv, Vb, 128, DD, DD, 0);
  regime_attn_kernel<<<32, 128, 0, stream>>>(Qb, Kb, Vb, stateb, alphag,
                                             rln_g, rln_b, (float*)d_out);
}